// Decoder_Attention_86552180949069
// MI455X (gfx1250) — compile-verified
//
#include <hip/hip_runtime.h>
#include <math.h>

// Problem constants (match reference)
#define B   32
#define T   16
#define D   128
#define HID 256
#define HW  576            // 24*24
#define G4  1024           // 4*HID

typedef __attribute__((ext_vector_type(2))) float v2f;
typedef __attribute__((ext_vector_type(8))) float v8f;

__device__ __forceinline__ float sigm(float x) { return 1.0f / (1.0f + expf(-x)); }

// ---------------------------------------------------------------------------
// Kernel 0: zero h/c state (harness poisons ws; we must init every launch)
// ---------------------------------------------------------------------------
__global__ __launch_bounds__(256) void init_state(float* __restrict__ h,
                                                  float* __restrict__ c) {
    int i = blockIdx.x * 256 + threadIdx.x;   // 2*B*HID = 16384 threads
    if (i < B * HID) h[i] = 0.0f;
    else             c[i - B * HID] = 0.0f;
}

// ---------------------------------------------------------------------------
// Kernel 1: h-independent attention logits.
// sx[t][hw][b] = sum_d dec_in[((t*B+b)*D+d)*HW + hw] * attn_w[d]
// One thread per (t,b,hw); consecutive threads = consecutive hw -> every
// d-iteration issues a fully coalesced 128B wave load. Streams all 151MB of
// dec_in from HBM (and leaves it resident in the 192MB L2 for the z pass).
// ---------------------------------------------------------------------------
__global__ __launch_bounds__(256) void sx_kernel(const float* __restrict__ x,
                                                 const float* __restrict__ attn_w,
                                                 float* __restrict__ sx) {
    __shared__ float wx[D];
    if (threadIdx.x < D) wx[threadIdx.x] = attn_w[threadIdx.x];
    __syncthreads();

    int id = blockIdx.x * 256 + threadIdx.x;   // T*B*HW = 294912 threads
    int hw = id % HW;
    int rest = id / HW;
    int b = rest & (B - 1);
    int t = rest >> 5;
    const float* xr = x + (size_t)((t * B + b) * D) * HW + hw;
    float acc = 0.0f;
#pragma unroll 8
    for (int d = 0; d < D; ++d) acc = fmaf(xr[(size_t)d * HW], wx[d], acc);
    sx[((size_t)t * HW + hw) * B + b] = acc;
}

// ---------------------------------------------------------------------------
// Kernel 2 (per step): sh[b] = h·w_h, then softmax over b per hw.
// Writes g to scratch ([b][hw], contiguous hw for the z pass) and to
// g_out[(b*HW+hw)*T + t] in d_out. Launch: 2 blocks x 288 threads (hw = id).
// ---------------------------------------------------------------------------
__global__ __launch_bounds__(288) void att_softmax_kernel(const float* __restrict__ sx,
                                                          const float* __restrict__ h,
                                                          const float* __restrict__ attn_w,
                                                          const float* __restrict__ attn_b,
                                                          float* __restrict__ g_scr,
                                                          float* __restrict__ g_out,
                                                          int t) {
    __shared__ float part[256];
    __shared__ float sh[B];
    int tid = threadIdx.x;

    // sh[b] = h[b,:]·attn_w[D:D+HID] + attn_b  (redundant per block; tiny)
    if (tid < 256) {
        int b = tid >> 3, seg = tid & 7;
        const float* hr = h + b * HID + seg * 32;
        const float* wr = attn_w + D + seg * 32;
        float p = 0.0f;
#pragma unroll
        for (int k = 0; k < 32; ++k) p = fmaf(hr[k], wr[k], p);
        part[tid] = p;
    }
    __syncthreads();
    if (tid < B) {
        float s = 0.0f;
#pragma unroll
        for (int i = 0; i < 8; ++i) s += part[tid * 8 + i];
        sh[tid] = s + attn_b[0];
    }
    __syncthreads();

    int hw = blockIdx.x * 288 + tid;           // 2*288 == HW exactly
    const float* sr = sx + ((size_t)t * HW + hw) * B;
    float v[B];
    float m = -1e30f;
#pragma unroll
    for (int b = 0; b < B; ++b) { v[b] = sr[b] + sh[b]; m = fmaxf(m, v[b]); }
    float sum = 0.0f;
#pragma unroll
    for (int b = 0; b < B; ++b) { v[b] = expf(v[b] - m); sum += v[b]; }
    float inv = 1.0f / sum;
#pragma unroll
    for (int b = 0; b < B; ++b) {
        float gv = v[b] * inv;
        g_scr[b * HW + hw]          = gv;
        g_out[(b * HW + hw) * T + t] = gv;   // g_out is (B,HW,T)
    }
}

// ---------------------------------------------------------------------------
// Kernel 3 (per step): z[b][d] = sum_hw x[t,b,d,hw] * g[b,hw].
// One wave per (b,d) row: 576-float contiguous row dotted with g (L2-hot x).
// ---------------------------------------------------------------------------
__global__ __launch_bounds__(256) void z_kernel(const float* __restrict__ x,
                                                const float* __restrict__ g_scr,
                                                float* __restrict__ z, int t) {
    int wave = threadIdx.x >> 5, lane = threadIdx.x & 31;
    int row = blockIdx.x * 8 + wave;           // 0..B*D-1 (4096)
    int b = row >> 7, d = row & (D - 1);
    const float* xr = x + (size_t)(((t * B + b) * D + d)) * HW;
    const float* gr = g_scr + b * HW;
    float acc = 0.0f;
#pragma unroll 2
    for (int hw = lane; hw < HW; hw += 32) acc = fmaf(xr[hw], gr[hw], acc);
#pragma unroll
    for (int off = 16; off > 0; off >>= 1) acc += __shfl_xor(acc, off, 32);
    if (lane == 0) z[row] = acc;
}

// ---------------------------------------------------------------------------
// Kernel 4 (per step): gates[32x1024] = z[32x128]·w_ih^T + h[32x256]·w_hh^T
// via V_WMMA_F32_16X16X4_F32 (exact fp32). 128 tiles (2 m-tiles x 64 n-tiles),
// one wave per tile, 96 k-steps. Fragment layouts per CDNA5 ISA 7.12.2:
//   A 16x4: lanes 0-15 hold {K=k0,k0+1}, lanes 16-31 hold {K=k0+2,k0+3}
//   B 4x16: dual (row k striped across lanes within a VGPR)
//   C/D: 8 VGPRs, VGPR v = row m0+v (lanes 0-15) / m0+v+8 (lanes 16-31)
// ---------------------------------------------------------------------------
__global__ __launch_bounds__(256) void gates_wmma_kernel(const float* __restrict__ z,
                                                         const float* __restrict__ h,
                                                         const float* __restrict__ w_ih,
                                                         const float* __restrict__ w_hh,
                                                         float* __restrict__ gates) {
    int lane = threadIdx.x & 31;
    int wave = threadIdx.x >> 5;
    int tile = blockIdx.x * 8 + wave;          // 0..127
    int m0 = (tile & 1) * 16;
    int n0 = (tile >> 1) * 16;
    int koff = (lane >> 4) * 2;                // 0 for lanes 0-15, 2 for 16-31
    int mr = m0 + (lane & 15);                 // A row this lane supplies
    int nc = n0 + (lane & 15);                 // B col this lane supplies

    v8f acc = {};
    // K = 0..127 : z x w_ih^T   (WB[k][n] = w_ih[n*128+k])
    for (int k0 = 0; k0 < D; k0 += 4) {
        int ka = k0 + koff;
        v2f a  = *(const v2f*)(z    + mr * D + ka);
        v2f bv = *(const v2f*)(w_ih + nc * D + ka);
        acc = __builtin_amdgcn_wmma_f32_16x16x4_f32(false, a, false, bv,
                                                    (short)0, acc, false, false);
    }
    // K = 128..383 : h x w_hh^T (WB[k][n] = w_hh[n*256+(k-128)])
    for (int k0 = 0; k0 < HID; k0 += 4) {
        int ka = k0 + koff;
        v2f a  = *(const v2f*)(h    + mr * HID + ka);
        v2f bv = *(const v2f*)(w_hh + nc * HID + ka);
        acc = __builtin_amdgcn_wmma_f32_16x16x4_f32(false, a, false, bv,
                                                    (short)0, acc, false, false);
    }
    int half = lane >> 4;
#pragma unroll
    for (int v = 0; v < 8; ++v)
        gates[(m0 + v + half * 8) * G4 + n0 + (lane & 15)] = acc[v];
}

// ---------------------------------------------------------------------------
// Kernel 5 (per step): biases + LSTM pointwise + conv_w accumulation.
// acc[b][j] += conv_w[t] * h_new ; at t==T-1 emit outp (B,1,HID) to d_out.
// ---------------------------------------------------------------------------
__global__ __launch_bounds__(256) void lstm_kernel(const float* __restrict__ gates,
                                                   const float* __restrict__ b_ih,
                                                   const float* __restrict__ b_hh,
                                                   const float* __restrict__ conv_w,
                                                   float* __restrict__ h,
                                                   float* __restrict__ c,
                                                   float* __restrict__ acc,
                                                   float* __restrict__ outp,
                                                   int t) {
    int idx = blockIdx.x * 256 + threadIdx.x;   // B*HID = 8192
    int b = idx >> 8, j = idx & (HID - 1);
    const float* gr = gates + b * G4;
    float gi = gr[j]           + b_ih[j]           + b_hh[j];
    float gf = gr[HID + j]     + b_ih[HID + j]     + b_hh[HID + j];
    float gg = gr[2 * HID + j] + b_ih[2 * HID + j] + b_hh[2 * HID + j];
    float go = gr[3 * HID + j] + b_ih[3 * HID + j] + b_hh[3 * HID + j];
    float cn = sigm(gf) * c[idx] + sigm(gi) * tanhf(gg);
    float hn = sigm(go) * tanhf(cn);
    c[idx] = cn;
    h[idx] = hn;
    float a = (t == 0 ? 0.0f : acc[idx]) + conv_w[t] * hn;
    acc[idx] = a;
    if (t == T - 1) outp[idx] = a;
}

// ---------------------------------------------------------------------------
extern "C" void kernel_launch(void* const* d_in, const int* in_sizes, int n_in,
                              void* d_out, int out_size, void* d_ws, size_t ws_size,
                              hipStream_t stream) {
    const float* dec_in = (const float*)d_in[0];
    const float* attn_w = (const float*)d_in[1];
    const float* attn_b = (const float*)d_in[2];
    const float* w_ih   = (const float*)d_in[3];
    const float* w_hh   = (const float*)d_in[4];
    const float* b_ih   = (const float*)d_in[5];
    const float* b_hh   = (const float*)d_in[6];
    const float* conv_w = (const float*)d_in[7];

    float* out  = (float*)d_out;
    float* outp  = out;            // (B,1,HID) = 8192 floats
    float* g_out = out + B * HID;  // (B,HW,T) = 294912 floats

    // workspace layout (floats) — total ~1.5 MB
    float* ws    = (float*)d_ws;
    float* sx    = ws;                       // T*HW*B  = 294912
    float* g_scr = sx    + (size_t)T * HW * B;  // B*HW = 18432
    float* zbuf  = g_scr + B * HW;           // B*D    = 4096
    float* hbuf  = zbuf  + B * D;            // B*HID  = 8192
    float* cbuf  = hbuf  + B * HID;          // B*HID  = 8192
    float* accb  = cbuf  + B * HID;          // B*HID  = 8192
    float* gates = accb  + B * HID;          // B*4HID = 32768

    init_state<<<(2 * B * HID) / 256, 256, 0, stream>>>(hbuf, cbuf);
    sx_kernel<<<(T * B * HW) / 256, 256, 0, stream>>>(dec_in, attn_w, sx);

    for (int t = 0; t < T; ++t) {
        att_softmax_kernel<<<2, 288, 0, stream>>>(sx, hbuf, attn_w, attn_b,
                                                  g_scr, g_out, t);
        z_kernel<<<(B * D) / 8, 256, 0, stream>>>(dec_in, g_scr, zbuf, t);
        gates_wmma_kernel<<<16, 256, 0, stream>>>(zbuf, hbuf, w_ih, w_hh, gates);
        lstm_kernel<<<(B * HID) / 256, 256, 0, stream>>>(gates, b_ih, b_hh, conv_w,
                                                         hbuf, cbuf, accb, outp, t);
    }
}